// TensorProduct_69295002354267
// MI455X (gfx1250) — compile-verified
//
#include <hip/hip_runtime.h>

typedef __attribute__((ext_vector_type(2))) float v2f;
typedef __attribute__((ext_vector_type(4))) float v4f;
typedef __attribute__((ext_vector_type(8))) float v8f;

#define ZDIM  32768
#define S0D   4
#define UD    32
#define S1D   4
#define VD    16
#define PD    16
#define SOUTD 8
#define ZT    32   // z rows per workgroup: two 16-row WMMA M-tiles (amortizes B loads)
#define NW    8    // waves per workgroup

#define G0S   (UD * 16 + 4)   // 516: +4 pad -> lane stride 4 mod 64 banks (conflict-free)
#define G1S   (VD * 16 + 4)   // 260

// Swizzled block-diagonal B table for v-pair-packed WMMA (K'=32, N=16 fully used):
//   k' = h*16 + k   (h = v parity within the pair, k = path)
//   n  = s + 8*h'   (s = output segment)
//   B'[k',n] = (h==h' && idx_out[k]==s) ? coeff[k,u,2vp+h] : 0
// Stored in exact per-lane fragment order: element ((uvp*4+g)*32+lane)*4+j holds
// B'[k'(i,lane), n(lane)] with i = g*4+j, k' = 4*(i>>1)+(i&1)+2*(lane>>4), n = lane&15,
// so each lane fetches its 16 B values with four coalesced b128 loads.
__device__ float g_Bsw[UD * (VD / 2) * 512];

__global__ void build_B_kernel(const float* __restrict__ coeff,
                               const int* __restrict__ idx_out) {
  int t = blockIdx.x * blockDim.x + threadIdx.x;   // (u, vp, lane)
  if (t >= UD * (VD / 2) * 32) return;
  int lane  = t & 31;
  int uvp   = t >> 5;                 // u*8 + vp
  int u     = uvp >> 3;
  int vp    = uvp & 7;
  int khalf = (lane >> 4) << 1;
  int n     = lane & 15;
  int sx    = n & 7;
  int hp    = n >> 3;
  for (int g = 0; g < 4; ++g) {
    for (int j = 0; j < 4; ++j) {
      int i  = g * 4 + j;
      int k2 = 4 * (i >> 1) + (i & 1) + khalf;     // k' in 0..31
      int h  = k2 >> 4;
      int k  = k2 & 15;                            // path index
      float val = (h == hp && idx_out[k] == sx)
                      ? coeff[k * (UD * VD) + u * VD + (2 * vp + h)]
                      : 0.0f;
      g_Bsw[((size_t)(uvp * 4 + g) * 32 + lane) * 4 + j] = val;
    }
  }
}

__device__ inline v8f wchain(v8f acc, v4f A0, v4f A1, v4f B0, v4f B1) {
  v2f a, b;
  a[0] = A0.x; a[1] = A0.y; b[0] = B0.x; b[1] = B0.y;
  acc = __builtin_amdgcn_wmma_f32_16x16x4_f32(false, a, false, b, (short)0, acc, false, false);
  a[0] = A0.z; a[1] = A0.w; b[0] = B0.z; b[1] = B0.w;
  acc = __builtin_amdgcn_wmma_f32_16x16x4_f32(false, a, false, b, (short)0, acc, false, false);
  a[0] = A1.x; a[1] = A1.y; b[0] = B1.x; b[1] = B1.y;
  acc = __builtin_amdgcn_wmma_f32_16x16x4_f32(false, a, false, b, (short)0, acc, false, false);
  a[0] = A1.z; a[1] = A1.w; b[0] = B1.z; b[1] = B1.w;
  acc = __builtin_amdgcn_wmma_f32_16x16x4_f32(false, a, false, b, (short)0, acc, false, false);
  return acc;
}

__global__ void __launch_bounds__(256)
tp_wmma_kernel(const float* __restrict__ x0, const float* __restrict__ x1,
               const int* __restrict__ idx0, const int* __restrict__ idx1,
               float* __restrict__ out) {
  __shared__ float g0s[ZT * G0S];                  // [z][u][slot16] (+pad)   ~66 KB
  __shared__ float g1s[ZT * G1S];                  // [z][v][slot16] (+pad)   ~33 KB
  __shared__ float stage[NW * ZT * SOUTD * VD];    // [w][z][s][v]            128 KB
  __shared__ int sIdx0[PD], sIdx1[PD];

  const int tid = threadIdx.x;
  const int z0  = blockIdx.x * ZT;

  if (tid < PD) { sIdx0[tid] = idx0[tid]; sIdx1[tid] = idx1[tid]; }
  __syncthreads();

  // Stage gathered inputs, already permuted into per-lane K-slot order.
  for (int i = tid; i < ZT * UD * 16; i += 256) {
    int z  = i >> 9;                 // /(UD*16)
    int r  = i & 511;
    int u  = r >> 4;
    int sl = r & 15;
    int p  = 4 * ((sl & 7) >> 1) + (sl & 1) + 2 * (sl >> 3);
    g0s[z * G0S + u * 16 + sl] =
        x0[(size_t)(z0 + z) * (S0D * UD) + sIdx0[p] * UD + u];
  }
  for (int i = tid; i < ZT * VD * 16; i += 256) {
    int z  = i >> 8;                 // /(VD*16)
    int r  = i & 255;
    int v  = r >> 4;
    int sl = r & 15;
    int p  = 4 * ((sl & 7) >> 1) + (sl & 1) + 2 * (sl >> 3);
    g1s[z * G1S + v * 16 + sl] =
        x1[(size_t)(z0 + z) * (S1D * VD) + sIdx1[p] * VD + v];
  }
  __syncthreads();

  const int lane  = tid & 31;
  const int wave  = tid >> 5;
  const int mrow  = lane & 15;          // A: M (=z)
  const int half8 = (lane >> 4) << 3;   // slot offset for this half-wave
  const int zhi   = (lane >> 4) << 3;
  const int sx    = lane & 7;           // D unpack: segment
  const int hv    = (lane >> 3) & 1;    // D unpack: v parity

  for (int uu = 0; uu < UD / NW; ++uu) {
    const int u = wave * (UD / NW) + uu;

    // Hoisted g0 factors for both M-tiles (z = mrow and z = mrow+16);
    // K-slot path pattern repeats every 16, so these serve both v's of a pair.
    const float* g0p = &g0s[mrow * G0S + u * 16 + half8];
    v4f G0a0 = *(const v4f*)(g0p);
    v4f G0a1 = *(const v4f*)(g0p + 4);
    v4f G0b0 = *(const v4f*)(g0p + 16 * G0S);
    v4f G0b1 = *(const v4f*)(g0p + 16 * G0S + 4);

    for (int vp = 0; vp < VD / 2; ++vp) {
      const int v0 = vp * 2;
      const float* g1p = &g1s[mrow * G1S + v0 * 16 + half8];
      v4f G1a0 = *(const v4f*)(g1p);                     // v0, tile 0
      v4f G1a1 = *(const v4f*)(g1p + 4);
      v4f G1a2 = *(const v4f*)(g1p + 16);                // v0+1, tile 0
      v4f G1a3 = *(const v4f*)(g1p + 20);
      v4f G1b0 = *(const v4f*)(g1p + 16 * G1S);          // v0, tile 1
      v4f G1b1 = *(const v4f*)(g1p + 16 * G1S + 4);
      v4f G1b2 = *(const v4f*)(g1p + 16 * G1S + 16);     // v0+1, tile 1
      v4f G1b3 = *(const v4f*)(g1p + 16 * G1S + 20);

      const float* Bp = &g_Bsw[(size_t)(u * (VD / 2) + vp) * 512 + lane * 4];
      v4f Bv0 = *(const v4f*)(Bp);
      v4f Bv1 = *(const v4f*)(Bp + 128);
      v4f Bv2 = *(const v4f*)(Bp + 256);
      v4f Bv3 = *(const v4f*)(Bp + 384);

      v4f Aa0 = G0a0 * G1a0, Aa1 = G0a1 * G1a1;   // K' 0..15  (h=0), tile 0
      v4f Aa2 = G0a0 * G1a2, Aa3 = G0a1 * G1a3;   // K' 16..31 (h=1), tile 0
      v4f Ab0 = G0b0 * G1b0, Ab1 = G0b1 * G1b1;   // tile 1
      v4f Ab2 = G0b0 * G1b2, Ab3 = G0b1 * G1b3;

      v8f acc0 = {}, acc1 = {};
      acc0 = wchain(acc0, Aa0, Aa1, Bv0, Bv1);    // 8x K=4 -> K'=32
      acc0 = wchain(acc0, Aa2, Aa3, Bv2, Bv3);
      acc1 = wchain(acc1, Ab0, Ab1, Bv0, Bv1);    // B fragments reused
      acc1 = wchain(acc1, Ab2, Ab3, Bv2, Bv3);

      // D: VGPR r, lane l -> z = r + 8*(l>>4); n = l&15 -> s = n&7, v = v0 + (n>>3).
      // Every element live: no predication.
      const int vcol = v0 + hv;
#pragma unroll
      for (int r = 0; r < 8; ++r) {
        stage[((wave * ZT + r + zhi) * SOUTD + sx) * VD + vcol]      = acc0[r];
        stage[((wave * ZT + r + zhi + 16) * SOUTD + sx) * VD + vcol] = acc1[r];
      }
    }

    // Flush this u's tile: 4096 floats per wave, b128 per lane,
    // 64B-contiguous per 4-lane group (v fastest in output layout).
    for (int it = 0; it < 32; ++it) {
      int flat = it * 32 + lane;            // over [z][s][vgroup]
      int vg   = (flat & 3) << 2;
      int rest = flat >> 2;
      int ss   = rest & 7;
      int z    = rest >> 3;
      v4f val  = *(const v4f*)&stage[((wave * ZT + z) * SOUTD + ss) * VD + vg];
      *(v4f*)&out[(size_t)(z0 + z) * (SOUTD * UD * VD) + ss * (UD * VD) + u * VD + vg] = val;
    }
  }
}

extern "C" void kernel_launch(void* const* d_in, const int* in_sizes, int n_in,
                              void* d_out, int out_size, void* d_ws, size_t ws_size,
                              hipStream_t stream) {
  (void)in_sizes; (void)n_in; (void)out_size; (void)d_ws; (void)ws_size;
  const float* x0      = (const float*)d_in[0];
  const float* x1      = (const float*)d_in[1];
  const float* coeff   = (const float*)d_in[2];
  const int*   idx0    = (const int*)d_in[3];
  const int*   idx1    = (const int*)d_in[4];
  const int*   idx_out = (const int*)d_in[5];
  float*       out     = (float*)d_out;

  build_B_kernel<<<(UD * (VD / 2) * 32 + 255) / 256, 256, 0, stream>>>(coeff, idx_out);
  tp_wmma_kernel<<<ZDIM / ZT, 256, 0, stream>>>(x0, x1, idx0, idx1, out);
}